// em_loss_63582695850902
// MI455X (gfx1250) — compile-verified
//
#include <hip/hip_runtime.h>
#include <math.h>

#define N_PTS 16384
#define DIM   64
#define KC    8
#define MAXIT 11
#define LOG2PI 1.837877066409345339081937709f
#define EPS32  1.1920928955078125e-07f     /* float32 eps */
#define EPSC   0.0009765625f               /* float16 eps (cov jitter) */

/* workspace layout, in floats */
#define WS_Q      0        /* 512    : 2*P_k*mu_k per k           */
#define WS_C      512      /* 8      : folded constant per k      */
#define WS_RED    520      /* 1      : lsl sum                    */
#define WS_NK     1016     /* 8      : sum of posteriors          */
#define WS_S1     1024     /* 512    : sum posterior*x            */
#define WS_P      2048     /* 32768  : precision matrices         */
#define WS_COVACC 34816    /* 32768  : covariance accumulators    */
#define WS_WL     67584    /* 131072 : weighted log-likelihood    */
#define WS_LSL    198656   /* 16384  : logsumexp per point        */
/* total 215040 floats ~= 860 KB */

typedef float v2f __attribute__((ext_vector_type(2)));
typedef float v8f __attribute__((ext_vector_type(8)));

/* =====================================================================
 * k_prep : per component k (8 blocks x 64 threads)
 *   it==0 : mu=mu_init, cov=I, pi=1/K
 *   it>0  : nkadj from Nk (any-zero EPS fixup), mu=S1/nkadj,
 *           cov = covacc/nkadj + eps*I   (fused former k_covfin/k_mufin)
 *   then Cholesky -> logdet -> L^-1 -> P = L^-T L^-1,
 *   q = 2*P*mu,  C_k = log(pi) - 0.5*(D*log2pi + logdet + mu^T P mu)
 *   zeroes its own covacc slice for this iteration.
 * ===================================================================== */
__global__ void __launch_bounds__(64) k_prep(const float* __restrict__ mu_init,
                                             float* __restrict__ ws, int it) {
    __shared__ float A[64][65];
    __shared__ float Li[64][65];
    __shared__ float mu_s[64];
    __shared__ float q_s[64];
    const int k = blockIdx.x;
    const int r = threadIdx.x;

    float lpi;
    if (it == 0) {
        mu_s[r] = mu_init[k * DIM + r];
        for (int c = 0; c < 64; ++c) A[r][c] = (r == c) ? 1.0f : 0.0f;
        lpi = logf(1.0f / (float)KC);
    } else {
        bool anyz = false;
        for (int j = 0; j < KC; ++j) anyz |= (ws[WS_NK + j] == 0.0f);
        float nka = ws[WS_NK + k] + (anyz ? EPS32 : 0.0f);
        float inv = 1.0f / nka;
        mu_s[r] = ws[WS_S1 + k * DIM + r] * inv;
        for (int c = 0; c < 64; ++c)
            A[r][c] = ws[WS_COVACC + k * DIM * DIM + r * 64 + c] * inv
                      + ((r == c) ? EPSC : 0.0f);
        lpi = logf(nka / (float)N_PTS);
    }
    __syncthreads();

    /* zero this component's covariance accumulator (block-local slice) */
    for (int j = 0; j < 64; ++j) ws[WS_COVACC + k * DIM * DIM + j * 64 + r] = 0.0f;

    /* Cholesky (lower), one thread per row */
    for (int j = 0; j < 64; ++j) {
        if (r == j) A[j][j] = sqrtf(A[j][j]);
        __syncthreads();
        if (r > j) A[r][j] /= A[j][j];
        __syncthreads();
        if (r > j) {
            float ljr = A[r][j];
            for (int c = j + 1; c <= r; ++c) A[r][c] -= ljr * A[c][j];
        }
        __syncthreads();
    }

    /* invert L: thread r solves column r by forward substitution */
    for (int i = 0; i < 64; ++i) Li[i][r] = 0.0f;
    Li[r][r] = 1.0f / A[r][r];
    for (int i = r + 1; i < 64; ++i) {
        float s = 0.0f;
        for (int m = r; m < i; ++m) s += A[i][m] * Li[m][r];
        Li[i][r] = -s / A[i][i];
    }
    __syncthreads();

    /* P row r, and q_r = (P*mu)_r */
    float qr = 0.0f;
    for (int e = 0; e < 64; ++e) {
        float s = 0.0f;
        int m0 = (r > e) ? r : e;
        for (int m = m0; m < 64; ++m) s += Li[m][r] * Li[m][e];
        ws[WS_P + k * DIM * DIM + r * 64 + e] = s;
        qr = fmaf(s, mu_s[e], qr);
    }
    ws[WS_Q + k * DIM + r] = 2.0f * qr;      /* pre-doubled */
    q_s[r] = qr;
    __syncthreads();
    if (r == 0) {
        float ld = 0.0f, cq = 0.0f;
        for (int i = 0; i < 64; ++i) {
            ld += logf(A[i][i]);
            cq = fmaf(q_s[i], mu_s[i], cq);
        }
        ws[WS_C + k] = lpi - 0.5f * ((float)DIM * LOG2PI + 2.0f * ld + cq);
    }
}

/* =====================================================================
 * k_estep : WMMA E-step + fused posterior accumulation.
 * Block = 64 points, 256 threads (8 waves).  Quadratic term T = X * P_k
 * via V_WMMA_F32_16X16X4_F32; X staged ONCE (k-independent A operand).
 *   wl[n,k] = -0.5 * sum_e (T[n,e] - 2q_e) x[n,e] + C_k
 * Then logsumexp, posterior, and S1/Nk block partials -> global atomics.
 * ===================================================================== */
__global__ void __launch_bounds__(256) k_estep(const float* __restrict__ x,
                                               float* __restrict__ ws) {
    __shared__ float Xb[64][68];   /* stride 68: conflict-free column reads */
    __shared__ float Pb[64][72];   /* stride 72: conflict-free row reads    */
    __shared__ float q2[64];
    __shared__ float mah[64];
    __shared__ float wlb[64][9];
    __shared__ float pst[64][9];

    const int tid   = threadIdx.x;
    const int nbase = blockIdx.x * 64;
    const int wv = tid >> 5, lane = tid & 31, lo = lane & 15, hi = lane >> 4;
    const int t0 = wv * 2;

    /* stage X tile once */
    for (int t = 0; t < 16; ++t) {
        int idx = t * 256 + tid;
        int p = idx >> 6, d = idx & 63;
        Xb[p][d] = x[(size_t)(nbase + p) * DIM + d];
    }
    __syncthreads();

    for (int k = 0; k < KC; ++k) {
        for (int t = 0; t < 16; ++t) {
            int idx = t * 256 + tid;
            int p = idx >> 6, d = idx & 63;
            Pb[p][d] = ws[WS_P + k * DIM * DIM + idx];
        }
        if (tid < 64) { q2[tid] = ws[WS_Q + k * DIM + tid]; mah[tid] = 0.0f; }
        __syncthreads();

        v8f acc[2];
#pragma unroll
        for (int ti = 0; ti < 2; ++ti)
#pragma unroll
            for (int r = 0; r < 8; ++r) acc[ti][r] = 0.0f;

#pragma unroll
        for (int dk = 0; dk < 16; ++dk) {
            const int d0 = dk * 4;
#pragma unroll
            for (int ti = 0; ti < 2; ++ti) {
                const int t  = t0 + ti;
                const int dt = t >> 2, et = t & 3;
                v2f a, b;
                a.x = Xb[dt * 16 + lo][d0 + 2 * hi];
                a.y = Xb[dt * 16 + lo][d0 + 2 * hi + 1];
                b.x = Pb[d0 + 2 * hi    ][et * 16 + lo];
                b.y = Pb[d0 + 2 * hi + 1][et * 16 + lo];
                acc[ti] = __builtin_amdgcn_wmma_f32_16x16x4_f32(
                    false, a, false, b, (short)0, acc[ti], false, false);
            }
        }

        /* reduce: mah[pt] += (T[pt][e] - 2q_e) * x[pt][e] */
#pragma unroll
        for (int ti = 0; ti < 2; ++ti) {
            const int t  = t0 + ti;
            const int dt = t >> 2, et = t & 3;
            const int e  = et * 16 + lo;
            const float qe = q2[e];
#pragma unroll
            for (int r = 0; r < 8; ++r) {
                const int pt = dt * 16 + r + 8 * hi;
                atomicAdd(&mah[pt], (acc[ti][r] - qe) * Xb[pt][e]);
            }
        }
        __syncthreads();

        if (tid < 64) {
            float wl = -0.5f * mah[tid] + ws[WS_C + k];
            wlb[tid][k] = wl;
            ws[WS_WL + (size_t)(nbase + tid) * KC + k] = wl;
        }
        __syncthreads();
    }

    /* logsumexp + posterior */
    if (tid < 64) {
        float m = wlb[tid][0];
#pragma unroll
        for (int j = 1; j < KC; ++j) m = fmaxf(m, wlb[tid][j]);
        float s = 0.0f;
#pragma unroll
        for (int j = 0; j < KC; ++j) s += expf(wlb[tid][j] - m);
        float lse = m + logf(s);
        ws[WS_LSL + nbase + tid] = lse;
#pragma unroll
        for (int j = 0; j < KC; ++j) pst[tid][j] = expf(wlb[tid][j] - lse);
    }
    __syncthreads();

    /* fused M-step partials: S1[k][d], Nk[k] */
    {
        const int d = tid & 63, g = tid >> 6;
        const int k0 = g, k1 = g + 4;
        float s0 = 0.0f, s1 = 0.0f;
        for (int p = 0; p < 64; ++p) {
            float xv = Xb[p][d];
            s0 = fmaf(pst[p][k0], xv, s0);
            s1 = fmaf(pst[p][k1], xv, s1);
        }
        atomicAdd(&ws[WS_S1 + k0 * DIM + d], s0);
        atomicAdd(&ws[WS_S1 + k1 * DIM + d], s1);
        if (tid < 8) {
            float sn = 0.0f;
            for (int p = 0; p < 64; ++p) sn += pst[p][tid];
            atomicAdd(&ws[WS_NK + tid], sn);
        }
    }
}

/* =====================================================================
 * k_cov : covariance contraction via V_WMMA_F32_16X16X4_F32.
 * mu_{new} recomputed locally from S1/Nk (former k_mufin folded in).
 * A = (w*dif2)^T [d x n], B = dif2 [n x e]; 8 waves x 2 tiles each.
 * ===================================================================== */
__global__ void __launch_bounds__(256) k_cov(const float* __restrict__ x,
                                             float* __restrict__ ws) {
    const int k = blockIdx.y;
    const int c = blockIdx.x;
    __shared__ float Ab[64][72];
    __shared__ float Bb[64][72];
    __shared__ float wbuf[64];
    __shared__ float muk[64];
    const int tid  = threadIdx.x;
    const int wv   = tid >> 5;
    const int lane = tid & 31;
    const int lo   = lane & 15;
    const int hi   = lane >> 4;

    if (tid < 64) {
        bool anyz = false;
        for (int j = 0; j < KC; ++j) anyz |= (ws[WS_NK + j] == 0.0f);
        float nka = ws[WS_NK + k] + (anyz ? EPS32 : 0.0f);
        muk[tid] = ws[WS_S1 + k * DIM + tid] / nka;
    }
    v8f acc[2];
#pragma unroll
    for (int ti = 0; ti < 2; ++ti)
#pragma unroll
        for (int r = 0; r < 8; ++r) acc[ti][r] = 0.0f;
    __syncthreads();

    for (int sc = 0; sc < 8; ++sc) {
        const int nbase = c * 512 + sc * 64;
        if (tid < 64) {
            int n = nbase + tid;
            wbuf[tid] = expf(ws[WS_WL + (size_t)n * KC + k] - ws[WS_LSL + n]);
        }
        __syncthreads();
        for (int tt = 0; tt < 16; ++tt) {
            int idx = tt * 256 + tid;
            int p = idx >> 6, d = idx & 63;
            int n = nbase + p;
            float dif = x[(size_t)n * DIM + d] - muk[d];
            Bb[p][d] = dif;
            Ab[p][d] = wbuf[p] * dif;
        }
        __syncthreads();
#pragma unroll
        for (int nk = 0; nk < 16; ++nk) {
            const int n0 = nk * 4;
#pragma unroll
            for (int ti = 0; ti < 2; ++ti) {
                const int t  = wv * 2 + ti;
                const int dt = t >> 2, et = t & 3;
                v2f a, b;
                a.x = Ab[n0 + 2 * hi    ][dt * 16 + lo];
                a.y = Ab[n0 + 2 * hi + 1][dt * 16 + lo];
                b.x = Bb[n0 + 2 * hi    ][et * 16 + lo];
                b.y = Bb[n0 + 2 * hi + 1][et * 16 + lo];
                acc[ti] = __builtin_amdgcn_wmma_f32_16x16x4_f32(
                    false, a, false, b, (short)0, acc[ti], false, false);
            }
        }
        __syncthreads();
    }

#pragma unroll
    for (int ti = 0; ti < 2; ++ti) {
        const int t  = wv * 2 + ti;
        const int dt = t >> 2, et = t & 3;
#pragma unroll
        for (int r = 0; r < 8; ++r) {
            int M  = dt * 16 + r + 8 * hi;
            int Nc = et * 16 + lo;
            atomicAdd(&ws[WS_COVACC + k * DIM * DIM + M * 64 + Nc], acc[ti][r]);
        }
    }
}

/* -------- outputs: one-hot(argmax wl), partial sum of lsl -------- */
__global__ void __launch_bounds__(256) k_out(float* __restrict__ out,
                                             float* __restrict__ ws) {
    const int n = blockIdx.x * 256 + threadIdx.x;
    const float4* wp = (const float4*)(ws + WS_WL + (size_t)n * KC);
    float4 a0 = wp[0], a1 = wp[1];
    float w[8] = {a0.x, a0.y, a0.z, a0.w, a1.x, a1.y, a1.z, a1.w};
    int am = 0; float bm = w[0];
#pragma unroll
    for (int j = 1; j < 8; ++j) if (w[j] > bm) { bm = w[j]; am = j; }
#pragma unroll
    for (int j = 0; j < 8; ++j)
        out[2 + (size_t)n * 8 + j] = (j == am) ? 1.0f : 0.0f;

    __shared__ float sd[256];
    sd[threadIdx.x] = ws[WS_LSL + n];
    __syncthreads();
    for (int s = 128; s > 0; s >>= 1) {
        if (threadIdx.x < s) sd[threadIdx.x] += sd[threadIdx.x + s];
        __syncthreads();
    }
    if (threadIdx.x == 0) atomicAdd(&ws[WS_RED], sd[0]);
}

__global__ void k_fin(float* __restrict__ out, const float* __restrict__ ws) {
    float m = ws[WS_RED] / (float)N_PTS;
    out[0] = -m;
    out[1] = m;
}

extern "C" void kernel_launch(void* const* d_in, const int* in_sizes, int n_in,
                              void* d_out, int out_size, void* d_ws, size_t ws_size,
                              hipStream_t stream) {
    (void)in_sizes; (void)n_in; (void)out_size; (void)ws_size;
    const float* x       = (const float*)d_in[0];
    const float* mu_init = (const float*)d_in[1];
    float* out = (float*)d_out;
    float* ws  = (float*)d_ws;

    hipMemsetAsync((char*)d_ws + WS_RED * sizeof(float), 0, sizeof(float), stream);

    for (int it = 0; it < MAXIT; ++it) {
        k_prep<<<KC, 64, 0, stream>>>(mu_init, ws, it);
        /* zero Nk+S1 (cross-block consumers forbid zeroing inside k_prep) */
        hipMemsetAsync((char*)d_ws + WS_NK * sizeof(float), 0,
                       (8 + 512) * sizeof(float), stream);
        k_estep<<<N_PTS / 64, 256, 0, stream>>>(x, ws);
        if (it != MAXIT - 1)   /* final iteration's M-step is dead */
            k_cov<<<dim3(N_PTS / 512, KC), 256, 0, stream>>>(x, ws);
    }

    k_out<<<N_PTS / 256, 256, 0, stream>>>(out, ws);
    k_fin<<<1, 1, 0, stream>>>(out, ws);
}